// GCN_15453292331332
// MI455X (gfx1250) — compile-verified
//
#include <hip/hip_runtime.h>

// GCN layer for MI455X (gfx1250):
//   deg/norm via f32 global atomics, edge scatter-add (memory bound, ~600K*128 atomics),
//   dense projection agg@W+b with V_WMMA_F32_16X16X4_F32 (exact f32 accumulation),
//   norm_dst folded into LDS A-tile staging, bias+ReLU fused into the store.

typedef __attribute__((ext_vector_type(2))) float v2f;
typedef __attribute__((ext_vector_type(8))) float v8f;

#define IN_FEATS  128
#define OUT_FEATS 128

// ---------------- zero workspace ----------------
__global__ void gcn_zero_kernel(float* __restrict__ p, long long n) {
  long long i = (long long)blockIdx.x * blockDim.x + threadIdx.x;
  long long stride = (long long)gridDim.x * blockDim.x;
  for (; i < n; i += stride) p[i] = 0.0f;
}

// ---------------- degree accumulation ----------------
__global__ void gcn_degree_kernel(const int* __restrict__ src,
                                  const int* __restrict__ dst,
                                  float* __restrict__ deg_out,
                                  float* __restrict__ deg_in,
                                  int n_edges) {
  int e = blockIdx.x * blockDim.x + threadIdx.x;
  if (e < n_edges) {
    atomicAdd(&deg_out[src[e]], 1.0f);
    atomicAdd(&deg_in[dst[e]], 1.0f);
  }
}

// ---------------- deg -> rsqrt(clamp(deg,1)) in place ----------------
__global__ void gcn_norm_kernel(float* __restrict__ deg, int n) {
  int i = blockIdx.x * blockDim.x + threadIdx.x;
  if (i < n) {
    float d = deg[i];
    deg[i] = rsqrtf(d < 1.0f ? 1.0f : d);
  }
}

// ---------------- edge gather / scatter-add ----------------
// One wave-lane group of 32 per edge; each lane moves a float4 (128 feats total).
__global__ void gcn_scatter_kernel(const float* __restrict__ feat,
                                   const int* __restrict__ src,
                                   const int* __restrict__ dst,
                                   const float* __restrict__ norm_src,
                                   float* __restrict__ agg,
                                   int n_edges) {
  long long tid = (long long)blockIdx.x * blockDim.x + threadIdx.x;
  int e    = (int)(tid >> 5);
  int lane = (int)(tid & 31);
  if (e >= n_edges) return;
  int s = src[e];
  int d = dst[e];
  float ns = norm_src[s];
  const float4* fs = (const float4*)(feat + (long long)s * IN_FEATS);
  float4 v = fs[lane];
  float* ad = agg + (long long)d * IN_FEATS + lane * 4;
  atomicAdd(ad + 0, v.x * ns);
  atomicAdd(ad + 1, v.y * ns);
  atomicAdd(ad + 2, v.z * ns);
  atomicAdd(ad + 3, v.w * ns);
}

// ---------------- WMMA GEMM: out = relu((agg * norm_dst[:,None]) @ W + b) ----------------
// Block = 256 threads = 8 waves. Block covers rows [m0, m0+16), each wave one
// 16-wide column tile (8 waves x 16 = 128 output cols). A staged in LDS with
// norm_dst scaling folded in; pad row stride to 132 floats to avoid bank conflicts.
__global__ void __launch_bounds__(256)
gcn_gemm_wmma_kernel(const float* __restrict__ agg,
                     const float* __restrict__ norm_dst,
                     const float* __restrict__ W,   // [128][128] row-major (K x N)
                     const float* __restrict__ bias,
                     float* __restrict__ out,
                     int n_nodes) {
  __shared__ float As[16][132];

  const int m0 = blockIdx.x * 16;
  const int t  = threadIdx.x;

  // Stage scaled A tile: 16 rows x 128 cols, 256 threads -> 8 floats each.
  for (int i = t; i < 16 * IN_FEATS; i += 256) {
    int r = i >> 7;          // row 0..15
    int c = i & 127;         // col 0..127
    int row = m0 + r;
    As[r][c] = agg[(long long)row * IN_FEATS + c] * norm_dst[row];
  }
  __syncthreads();

  const int wave = t >> 5;
  const int lane = t & 31;
  const int n0   = wave * 16;
  const int half = lane >> 4;   // 0: lanes 0-15, 1: lanes 16-31
  const int l    = lane & 15;

  v8f c = {};
  #pragma unroll
  for (int k0 = 0; k0 < IN_FEATS; k0 += 4) {
    // A 16x4 f32 layout: lanes 0-15 hold M=l, K={k0,k0+1}; lanes 16-31 K={k0+2,k0+3}
    v2f a;
    a.x = As[l][k0 + half * 2 + 0];
    a.y = As[l][k0 + half * 2 + 1];
    // B 4x16 f32 layout (symmetric): lane holds N=l, K={k0(+2),k0+1(+3)}
    v2f bm;
    bm.x = W[(k0 + half * 2 + 0) * OUT_FEATS + n0 + l];
    bm.y = W[(k0 + half * 2 + 1) * OUT_FEATS + n0 + l];
    // 8 args: (neg_a, A, neg_b, B, c_mod, C, reuse_a, reuse_b)
    c = __builtin_amdgcn_wmma_f32_16x16x4_f32(false, a, false, bm,
                                              (short)0, c, false, false);
  }

  const float bv = bias[n0 + l];
  #pragma unroll
  for (int r = 0; r < 8; r++) {
    // C/D layout: VGPR r -> M = r (lanes 0-15) or M = r+8 (lanes 16-31); N = l
    int row = m0 + r + half * 8;
    float v = c[r] + bv;
    out[(long long)row * OUT_FEATS + n0 + l] = v > 0.0f ? v : 0.0f;
  }
}

extern "C" void kernel_launch(void* const* d_in, const int* in_sizes, int n_in,
                              void* d_out, int out_size, void* d_ws, size_t ws_size,
                              hipStream_t stream) {
  const float* feat = (const float*)d_in[0];
  const int*   src  = (const int*)d_in[1];   // harness convention: integer -> int32
  const int*   dst  = (const int*)d_in[2];
  const float* W    = (const float*)d_in[3];
  const float* bias = (const float*)d_in[4];
  float* out = (float*)d_out;

  const int n_nodes = in_sizes[0] / IN_FEATS;   // 50000
  const int n_edges = in_sizes[1];              // 600000

  // Workspace layout: deg_out[N] | deg_in[N] | agg[N*128]
  float* deg_out = (float*)d_ws;
  float* deg_in  = deg_out + n_nodes;
  float* agg     = deg_in + n_nodes;

  // 1) zero degrees + agg (contiguous region)
  long long zero_n = (long long)n_nodes * (2 + IN_FEATS);
  gcn_zero_kernel<<<2048, 256, 0, stream>>>(deg_out, zero_n);

  // 2) degrees
  gcn_degree_kernel<<<(n_edges + 255) / 256, 256, 0, stream>>>(
      src, dst, deg_out, deg_in, n_edges);

  // 3) rsqrt norms in place (both arrays are contiguous: 2*n_nodes)
  gcn_norm_kernel<<<(2 * n_nodes + 255) / 256, 256, 0, stream>>>(
      deg_out, 2 * n_nodes);

  // 4) edge gather/scale/scatter-add: 32 lanes per edge
  {
    long long threads = (long long)n_edges * 32;
    int blocks = (int)((threads + 255) / 256);
    gcn_scatter_kernel<<<blocks, 256, 0, stream>>>(
        feat, src, dst, /*norm_src=*/deg_out, agg, n_edges);
  }

  // 5) WMMA GEMM + bias + relu (n_nodes = 3125 * 16, exact tiling)
  gcn_gemm_wmma_kernel<<<n_nodes / 16, 256, 0, stream>>>(
      agg, /*norm_dst=*/deg_in, W, bias, out, n_nodes);
}